// BrushStrokeRenderer_22393959481504
// MI455X (gfx1250) — compile-verified
//
#include <hip/hip_runtime.h>
#include <stdint.h>

#define Hh 512
#define Ww 512
#define NS 5000
#define SPTS 10
#define KNEAR 20
#define HC 102
#define WC 102
#define NCELL (HC*WC)      /* 10404 */
#define NS_PAD 5008
#define NTILES 313         /* ceil(5000/16) */

/* ws layout (bytes):
   pts     @ 0        : NS*SPTS*2 f32 = 400000
   stroke2 @ 400000   : NS_PAD f32    = 20032
   idcs    @ 420096   : NCELL*K int   = 832320
   wcoarse @ 1252416  : NCELL*K f32   = 832320   (total ~2.0 MB) */
#define WS_PTS 0
#define WS_S2  400000
#define WS_IDX 420096
#define WS_WC  1252416

typedef float v2f __attribute__((ext_vector_type(2)));
typedef float v8f __attribute__((ext_vector_type(8)));

__device__ __forceinline__ int imax(int a, int b) { return a > b ? a : b; }
__device__ __forceinline__ int imin(int a, int b) { return a < b ? a : b; }

/* ---------------- Kernel 1: bezier points + |loc|^2 ---------------- */
__global__ void k_prep(const float* __restrict__ cs, const float* __restrict__ ce,
                       const float* __restrict__ cc, const float* __restrict__ loc,
                       float* __restrict__ pts, float* __restrict__ s2) {
  int tid = blockIdx.x * blockDim.x + threadIdx.x;
  if (tid < NS * SPTS) {
    int n = tid / SPTS, s = tid % SPTS;
    float lx = loc[2*n], ly = loc[2*n+1];
    float sx = cs[2*n] + lx, sy = cs[2*n+1] + ly;
    float ex = ce[2*n] + lx, ey = ce[2*n+1] + ly;
    float px = cc[2*n] + lx, py = cc[2*n+1] + ly;
    float t = (float)s * (1.0f/9.0f);
    float omt2 = (1.0f - t) * (1.0f - t), t2 = t * t;
    pts[tid*2+0] = px + omt2*(sx - px) + t2*(ex - px);
    pts[tid*2+1] = py + omt2*(sy - py) + t2*(ey - py);
  }
  if (tid < NS_PAD) {
    float v = 0.0f;
    if (tid < NS) { float lx = loc[2*tid], ly = loc[2*tid+1]; v = lx*lx + ly*ly; }
    s2[tid] = v;
  }
}

/* ------- Kernel 2: coarse top-K via WMMA distance cross-term -------
   One wave = 16 coarse cells. A = stroke coords (M=16,K=4), B = cell coords
   (K=4,N=16).  D[m][n] lands with cell n in lane (n, n+16) and stroke m in
   VGPR m%8, so each lane keeps a private register top-20 over half the
   strokes; halves merge through LDS.  Candidate validity is uniform per
   (tile, half) because NS is a multiple of 8, so a single hoisted guard
   covers the unrolled 8-candidate insertion block. */
__global__ void __launch_bounds__(32) k_topk(const float* __restrict__ loc,
                                             const float* __restrict__ s2,
                                             const float* __restrict__ width,
                                             int* __restrict__ idcs,
                                             float* __restrict__ wc) {
  __shared__ unsigned long long lk[16 * 40];
  const int lane = threadIdx.x;
  const int c    = lane & 15;
  const int half = lane >> 4;
  const int cell = blockIdx.x * 16 + c;
  const bool cellOk = cell < NCELL;
  int cy = cell / WC, cx = cell % WC;
  float cellX = cellOk ? (float)cy * (512.0f/101.0f) : 0.0f;   /* linspace(0,512,102) */
  float cellY = cellOk ? (float)cx * (512.0f/101.0f) : 0.0f;
  float cell2 = cellX*cellX + cellY*cellY;
  v2f bmat;
  bmat.x = (half == 0) ? cellX : 0.0f;   /* lanes 16-31 carry K=2,3 (zero pad) */
  bmat.y = (half == 0) ? cellY : 0.0f;

  unsigned long long keys[KNEAR];
#pragma unroll
  for (int i = 0; i < KNEAR; ++i) keys[i] = ~0ull;
  unsigned long long worst = ~0ull;
  int worstSlot = 0;

  for (int t = 0; t < NTILES; ++t) {
    int base = t * 16;
    v2f amat; amat.x = 0.0f; amat.y = 0.0f;
    int ms = base + lane;
    if (half == 0 && ms < NS) { amat.x = loc[2*ms]; amat.y = loc[2*ms+1]; }
    v8f acc = {0.f,0.f,0.f,0.f,0.f,0.f,0.f,0.f};
    acc = __builtin_amdgcn_wmma_f32_16x16x4_f32(false, amat, false, bmat,
                                                (short)0, acc, false, false);
    const int mbase = base + 8*half;
    const float4* s2p = (const float4*)(s2 + mbase);   /* padded: always in-bounds */
    float4 q0 = s2p[0], q1 = s2p[1];
    float s2v[8] = {q0.x,q0.y,q0.z,q0.w,q1.x,q1.y,q1.z,q1.w};
    if (mbase < NS) {            /* uniform per (tile,half): one exec guard */
#pragma unroll
      for (int j = 0; j < 8; ++j) {
        float dist = fmaxf(cell2 + s2v[j] - 2.0f*acc[j], 0.0f);
        unsigned long long key =
            (((unsigned long long)__float_as_uint(dist)) << 13) |
            (unsigned)(mbase + j);
        if (key < worst) {
#pragma unroll
          for (int s = 0; s < KNEAR; ++s) if (s == worstSlot) keys[s] = key;
          worst = keys[0]; worstSlot = 0;
#pragma unroll
          for (int s = 1; s < KNEAR; ++s)
            if (keys[s] > worst) { worst = keys[s]; worstSlot = s; }
        }
      }
    }
  }
#pragma unroll
  for (int s = 0; s < KNEAR; ++s) lk[c*40 + half*20 + s] = keys[s];
  __syncthreads();
  if (half == 0 && cellOk) {
    unsigned long long last = 0ull;
    for (int r = 0; r < KNEAR; ++r) {
      unsigned long long best = ~0ull;
      for (int j = 0; j < 40; ++j) {
        unsigned long long k2 = lk[c*40 + j];
        if ((r == 0 || k2 > last) && k2 < best) best = k2;
      }
      last = best;
      int idx = (int)(best & 8191ull);
      idcs[cell*KNEAR + r] = idx;
      wc[cell*KNEAR + r]   = width[idx];
    }
  }
}

/* ---------- Kernel 3: full-res render, async LDS staging ----------- */
__global__ void __launch_bounds__(64) k_render(const float* __restrict__ color,
                                               const float* __restrict__ wcoarse,
                                               const int* __restrict__ idcs,
                                               const float* __restrict__ pts,
                                               float* __restrict__ out) {
  __shared__ int   sIdx[KNEAR];
  __shared__ float sPts[KNEAR * SPTS * 2] __attribute__((aligned(16)));
  __shared__ float sCol[KNEAR * 3];
  __shared__ float sW[9 * KNEAR];
  const int cell = blockIdx.x;
  const int cy = cell / WC, cx = cell % WC;
  const int t = threadIdx.x;

  if (t < KNEAR) sIdx[t] = idcs[cell*KNEAR + t];
  __syncthreads();

  /* async global -> LDS staging (ASYNCcnt path) */
  unsigned ptsLds = (unsigned)(uintptr_t)(&sPts[0]);
  for (int cch = t; cch < 100; cch += 64) {             /* 20 strokes x 5 x 16B */
    int k = cch / 5, p = cch % 5;
    unsigned long long src = (unsigned long long)(uintptr_t)
        ((const char*)pts + (size_t)sIdx[k]*80 + (size_t)p*16);
    unsigned dst = ptsLds + cch*16;
    asm volatile("global_load_async_to_lds_b128 %0, %1, off"
                 :: "v"(dst), "v"(src) : "memory");
  }
  unsigned colLds = (unsigned)(uintptr_t)(&sCol[0]);
  if (t < 60) {
    int k = t / 3, f = t % 3;
    unsigned long long src = (unsigned long long)(uintptr_t)
        ((const char*)color + ((size_t)sIdx[k]*3 + f)*4);
    unsigned dst = colLds + t*4;
    asm volatile("global_load_async_to_lds_b32 %0, %1, off"
                 :: "v"(dst), "v"(src) : "memory");
  }
  unsigned wLds = (unsigned)(uintptr_t)(&sW[0]);
  for (int e = t; e < 9*KNEAR; e += 64) {               /* 3x3 clamped neighbor widths */
    int cslot = e / KNEAR, k = e % KNEAR;
    int ccy = imin(imax(cy - 1 + cslot/3, 0), HC-1);
    int ccx = imin(imax(cx - 1 + cslot%3, 0), WC-1);
    unsigned long long src = (unsigned long long)(uintptr_t)
        ((const char*)wcoarse + ((size_t)(ccy*WC + ccx)*KNEAR + k)*4);
    unsigned dst = wLds + e*4;
    asm volatile("global_load_async_to_lds_b32 %0, %1, off"
                 :: "v"(dst), "v"(src) : "memory");
  }
  asm volatile("s_wait_asynccnt 0x0" ::: "memory");
  __syncthreads();

  /* pixel range of this coarse cell: floor(y*51/256)==cy */
  int yS = (256*cy + 50) / 51, yE = (256*cy + 306) / 51;
  int xS = (256*cx + 50) / 51, xE = (256*cx + 306) / 51;
  int nx = xE - xS, ny = yE - yS;
  if (t >= nx * ny) return;
  int py = yS + t / nx, px = xS + t % nx;
  float p0 = (float)py * (512.0f/511.0f);    /* linspace(0,512,512) */
  float p1 = (float)px * (512.0f/511.0f);

  float mps[KNEAR];
  float Dmin = 3.4e38f;
#pragma unroll
  for (int k = 0; k < KNEAR; ++k) {
    const float* pk = &sPts[k * SPTS * 2];
    float best = 3.4e38f;
    float ax = pk[0], ay = pk[1];
#pragma unroll
    for (int s = 0; s < SPTS - 1; ++s) {
      float bx = pk[2*s+2], by = pk[2*s+3];
      float dx = bx - ax, dy = by - ay;
      float pax = p0 - ax, pay = p1 - ay;
      float tt = (dx*pax + dy*pay) / (dx*dx + dy*dy);
      tt = fminf(fmaxf(tt, 0.0f), 1.0f);
      float qx = ax + tt*dx, qy = ay + tt*dy;
      float ex = p0 - qx, ey = p1 - qy;
      best = fminf(best, ex*ex + ey*ey);
      ax = bx; ay = by;
    }
    mps[k] = best;
    Dmin = fminf(Dmin, best);
  }
  float Lmax = 100000.0f / (1e-8f + Dmin);   /* logit monotone-decreasing in mps */

  /* bilinear (jax.image.resize 'linear', half-pixel centers, edge clamp) */
  float fy = ((float)py + 0.5f) * (102.0f/512.0f) - 0.5f;
  float fx = ((float)px + 0.5f) * (102.0f/512.0f) - 0.5f;
  float fy0 = floorf(fy), fx0 = floorf(fx);
  float wy = fy - fy0, wx = fx - fx0;
  int ly0 = (int)fy0 - cy + 1;               /* 0 or 1 into staged 3x3 */
  int lx0 = (int)fx0 - cx + 1;
  const float* W00 = &sW[(ly0*3 + lx0    )*KNEAR];
  const float* W01 = &sW[(ly0*3 + lx0 + 1)*KNEAR];
  const float* W10 = &sW[((ly0+1)*3 + lx0    )*KNEAR];
  const float* W11 = &sW[((ly0+1)*3 + lx0 + 1)*KNEAR];
  float w00 = (1.f-wy)*(1.f-wx), w01 = (1.f-wy)*wx, w10 = wy*(1.f-wx), w11 = wy*wx;

  float Z = 0.f, aR = 0.f, aG = 0.f, aB = 0.f, aW = 0.f;
#pragma unroll
  for (int k = 0; k < KNEAR; ++k) {
    float l = 100000.0f / (1e-8f + mps[k]);
    float e = expf(l - Lmax);
    Z  += e;
    aR += e * sCol[k*3+0];
    aG += e * sCol[k*3+1];
    aB += e * sCol[k*3+2];
    float wk = w00*W00[k] + w01*W01[k] + w10*W10[k] + w11*W11[k];
    aW += e * wk;
  }
  float inv = 1.0f / Z;
  float bs = aW * inv;
  float mask = 1.0f / (1.0f + expf(-(bs - Dmin)));
  float om = (1.0f - mask) * 0.5f;
  size_t o = ((size_t)py * Ww + px) * 3;
  out[o+0] = aR*inv*mask + om;
  out[o+1] = aG*inv*mask + om;
  out[o+2] = aB*inv*mask + om;
}

extern "C" void kernel_launch(void* const* d_in, const int* in_sizes, int n_in,
                              void* d_out, int out_size, void* d_ws, size_t ws_size,
                              hipStream_t stream) {
  const float* cs    = (const float*)d_in[0];
  const float* ce    = (const float*)d_in[1];
  const float* cc    = (const float*)d_in[2];
  const float* color = (const float*)d_in[3];
  const float* loc   = (const float*)d_in[4];
  const float* width = (const float*)d_in[5];
  float* out = (float*)d_out;
  char* ws = (char*)d_ws;
  float* pts = (float*)(ws + WS_PTS);
  float* s2  = (float*)(ws + WS_S2);
  int*   idx = (int*)  (ws + WS_IDX);
  float* wc  = (float*)(ws + WS_WC);
  (void)in_sizes; (void)n_in; (void)out_size; (void)ws_size;

  k_prep <<<(NS*SPTS + 255)/256, 256, 0, stream>>>(cs, ce, cc, loc, pts, s2);
  k_topk <<<(NCELL + 15)/16,      32, 0, stream>>>(loc, s2, width, idx, wc);
  k_render<<<NCELL,               64, 0, stream>>>(color, wc, idx, pts, out);
}